// SNNCell_7722351199143
// MI455X (gfx1250) — compile-verified
//
#include <hip/hip_runtime.h>

typedef __attribute__((ext_vector_type(2))) float v2f;
typedef __attribute__((ext_vector_type(8))) float v8f;

#define BETA    0.99f
#define THRESH  1.0f
#define NSTEP   50
#define K_DIM   512
#define N_DIM   256
#define M_DIM   8192

// Block: 256 threads = 8 waves (wave32), arranged 2 (M) x 4 (N) of 16x16 WMMA tiles.
// Block tile: 32 rows (batch) x 64 cols (neurons). Grid: (8192/32)*(256/64) = 1024.
__global__ __launch_bounds__(256)
void snn_leaky_wmma_kernel(const float* __restrict__ x,
                           const float* __restrict__ W,
                           const float* __restrict__ bias,
                           float* __restrict__ out) {
    const int tid  = threadIdx.x;
    const int wave = tid >> 5;
    const int lane = tid & 31;
    const int half = lane >> 4;   // 0: lanes 0-15, 1: lanes 16-31
    const int l16  = lane & 15;

    const int wm = wave >> 2;     // 0..1  (M sub-tile)
    const int wn = wave & 3;      // 0..3  (N sub-tile)

    const int ntile = blockIdx.x & 3;
    const int mtile = blockIdx.x >> 2;
    const int Mbase = mtile * 32 + wm * 16;
    const int Nbase = ntile * 64 + wn * 16;

    // A fragment (16x4 f32): lane holds row M=l16, K pair (2*half, 2*half+1) -> contiguous float2.
    // B fragment (4x16 f32): lane holds col N=l16, K pair (2*half, 2*half+1); B[k][n] = W[n][k].
    const float* xp = x + (size_t)(Mbase + l16) * K_DIM + 2 * half;
    const float* wp = W + (size_t)(Nbase + l16) * K_DIM + 2 * half;

    v8f acc = {};
#pragma unroll 8
    for (int k = 0; k < K_DIM; k += 4) {
        v2f a = *(const v2f*)(xp + k);
        v2f b = *(const v2f*)(wp + k);
        // D = A(16x4) * B(4x16) + C, f32 throughout (exact vs f32 reference)
        acc = __builtin_amdgcn_wmma_f32_16x16x4_f32(
            /*neg_a=*/false, a, /*neg_b=*/false, b,
            /*c_mod=*/(short)0, acc, /*reuse_a=*/false, /*reuse_b=*/false);
    }

    // C/D layout: VGPR i, lanes 0-15 -> M=i, lanes 16-31 -> M=i+8; N = l16 for all 8 VGPRs.
    const int   ncol = Nbase + l16;
    const float bn   = bias[ncol];

#pragma unroll
    for (int i = 0; i < 8; ++i) {
        const float cur = acc[i] + bn;
        const int   mrow = Mbase + i + 8 * half;
        float* op = out + ((size_t)mrow * N_DIM + ncol) * NSTEP;

        // Leaky integrate-and-fire, reset_mechanism='subtract':
        //   reset = H(mem - 1)  (membrane BEFORE update)
        //   mem   = beta*mem + cur - reset*1
        //   spk   = H(mem - 1)
        float mem = 0.0f;
#pragma unroll
        for (int t = 0; t < NSTEP; t += 2) {
            float r0 = (mem > THRESH) ? THRESH : 0.0f;
            mem = __builtin_fmaf(BETA, mem, cur - r0);
            float s0 = (mem > THRESH) ? 1.0f : 0.0f;

            float r1 = s0 * THRESH;          // reset for next step == spike just fired
            mem = __builtin_fmaf(BETA, mem, cur - r1);
            float s1 = (mem > THRESH) ? 1.0f : 0.0f;

            v2f sv = { s0, s1 };
            *(v2f*)(op + t) = sv;            // 8B-aligned contiguous store along T
        }
    }
}

extern "C" void kernel_launch(void* const* d_in, const int* in_sizes, int n_in,
                              void* d_out, int out_size, void* d_ws, size_t ws_size,
                              hipStream_t stream) {
    const float* x  = (const float*)d_in[0];   // [8192, 512]
    const float* W  = (const float*)d_in[1];   // [256, 512]
    const float* b  = (const float*)d_in[2];   // [256]
    float* out      = (float*)d_out;           // [8192, 256, 50]

    const int grid = (M_DIM / 32) * (N_DIM / 64);   // 1024 blocks
    snn_leaky_wmma_kernel<<<grid, 256, 0, stream>>>(x, W, b, out);
}